// MoELayer_34711925686738
// MI455X (gfx1250) — compile-verified
//
#include <hip/hip_runtime.h>
#include <stdint.h>

typedef float v2f __attribute__((ext_vector_type(2)));
typedef float v4f __attribute__((ext_vector_type(4)));
typedef float v8f __attribute__((ext_vector_type(8)));

namespace {
constexpr int kBc = 2, kSc = 2048, kD = 1024, kH = 2048, kE = 8;
constexpr int kM = kBc * kSc;              // 4096 tokens
constexpr float kScale = 1.0f / 32.0f;     // 1/sqrt(1024)

constexpr int BM = 128, BN = 128, BK = 16, PAD = 4;
constexpr int BNP = 144;  // B LDS n-stride; 2*BNP mod 64 == 32 -> halves of the
                          // wave hit opposite 32-bank groups (conflict-free)
} // namespace

// ---- CDNA5 async global->LDS copy (ASYNCcnt-tracked, bypasses VGPRs) ------
__device__ __forceinline__ void async_copy_b128(uint32_t lds_byte_addr,
                                                const void* gaddr) {
  asm volatile("global_load_async_to_lds_b128 %0, %1, off"
               :
               : "v"(lds_byte_addr), "v"((unsigned long long)(uintptr_t)gaddr)
               : "memory");
}
__device__ __forceinline__ void wait_async0() {
  asm volatile("s_wait_asynccnt 0x0" ::: "memory");
}
// low 32 bits of a generic pointer to __shared__ = LDS byte offset
__device__ __forceinline__ uint32_t lds_addr32(const void* p) {
  return (uint32_t)(uintptr_t)p;
}

// ---------------------------------------------------------------------------
// Router: one wave32 per token. Dense [token, E] weight map in workspace.
// ---------------------------------------------------------------------------
__global__ __launch_bounds__(256) void router_kernel(
    const float* __restrict__ x,    // [kM, kD]
    const float* __restrict__ Wr,   // [kD, kE]
    const float* __restrict__ rb,   // [kE]
    float* __restrict__ wfull) {    // [kM, kE]
  const int wave = threadIdx.x >> 5;
  const int lane = threadIdx.x & 31;
  const int token = blockIdx.x * 8 + wave;
  const float* xr = x + (size_t)token * kD;

  float acc[kE];
#pragma unroll
  for (int e = 0; e < kE; ++e) acc[e] = 0.0f;

  for (int d = lane; d < kD; d += 32) {
    const float xv = xr[d];
    const float* wr = Wr + (size_t)d * kE;
#pragma unroll
    for (int e = 0; e < kE; ++e) acc[e] = fmaf(xv, wr[e], acc[e]);
  }
#pragma unroll
  for (int e = 0; e < kE; ++e) {
#pragma unroll
    for (int off = 16; off >= 1; off >>= 1)
      acc[e] += __shfl_xor(acc[e], off, 32);
  }

  if (lane == 0) {
    float logit[kE], biased[kE];
#pragma unroll
    for (int e = 0; e < kE; ++e) {
      logit[e] = acc[e] * kScale;
      biased[e] = logit[e] + rb[e];
    }
    // top-2 on biased logits, lowest-index tie-break (matches lax.top_k)
    int i0 = 0;
#pragma unroll
    for (int e = 1; e < kE; ++e)
      if (biased[e] > biased[i0]) i0 = e;
    int i1 = (i0 == 0) ? 1 : 0;
#pragma unroll
    for (int e = 0; e < kE; ++e)
      if (e != i0 && biased[e] > biased[i1]) i1 = e;
    // softmax over the two ORIGINAL logits
    const float l0 = logit[i0], l1 = logit[i1];
    const float mx = fmaxf(l0, l1);
    const float e0 = __expf(l0 - mx), e1 = __expf(l1 - mx);
    const float inv = 1.0f / (e0 + e1);
    const float w0 = e0 * inv, w1 = e1 * inv;
    float* wf = wfull + (size_t)token * kE;
#pragma unroll
    for (int e = 0; e < kE; ++e)
      wf[e] = (e == i0) ? w0 : (e == i1) ? w1 : 0.0f;
  }
}

// ---------------------------------------------------------------------------
// Pass 1: hbuf = (x @ W1) * silu(x @ W3)     M=4096, N=kH, K=kD
// Block tile 128x128x16; 8 waves (4 M x 2 N); wave tile 32x64 (2x4 16x16).
// A tile: double-buffered LDS, filled by GLOBAL_LOAD_ASYNC_TO_LDS_B128.
// B tiles: reg-staged into k-paired LDS layout [kpair][n][2] so every
// fragment is one aligned ds_load_b64 (no register shuffles before WMMA).
// ---------------------------------------------------------------------------
__global__ __launch_bounds__(256) void gateup_kernel(
    const float* __restrict__ A,    // [kM, kD]
    const float* __restrict__ B1,   // [kD, kH]
    const float* __restrict__ B3,   // [kD, kH]
    float* __restrict__ Hout) {     // [kM, kH]
  __shared__ __align__(16) float sA[2][BM][BK + PAD];
  __shared__ __align__(16) float sB1[BK / 2][BNP][2];
  __shared__ __align__(16) float sB3[BK / 2][BNP][2];

  const int m0 = blockIdx.y * BM;
  const int n0 = blockIdx.x * BN;
  const int tid = threadIdx.x;
  const int wave = tid >> 5;
  const int lane = tid & 31;
  const int wm = (wave & 3) * 32;    // wave M offset: 0,32,64,96
  const int wn = (wave >> 2) * 64;   // wave N offset: 0,64
  const int r16 = lane & 15;
  const int koff = (lane >> 4) * 2;  // upper half-wave holds k+2,k+3
  const int rowb = (lane >> 4) * 8;  // C/D: upper half-wave holds rows +8

  // staging-thread mappings
  const int aRow0 = tid >> 2;        // A: thread -> (row, 16B chunk), x2
  const int aC4_0 = (tid & 3) * 4;
  const int aRow1 = (tid + 256) >> 2;
  const int aC4_1 = ((tid + 256) & 3) * 4;
  const int bKp = tid >> 5;          // B: k-pair 0..7
  const int bC4 = (tid & 31) * 4;    // B: n offset within tile

  v8f acc1[2][4], acc3[2][4];
#pragma unroll
  for (int mt = 0; mt < 2; ++mt)
#pragma unroll
    for (int nt = 0; nt < 4; ++nt) {
      acc1[mt][nt] = (v8f){0, 0, 0, 0, 0, 0, 0, 0};
      acc3[mt][nt] = (v8f){0, 0, 0, 0, 0, 0, 0, 0};
    }

  auto issueA = [&](int buf, int k0) {
    async_copy_b128(lds_addr32(&sA[buf][aRow0][aC4_0]),
                    A + (size_t)(m0 + aRow0) * kD + k0 + aC4_0);
    async_copy_b128(lds_addr32(&sA[buf][aRow1][aC4_1]),
                    A + (size_t)(m0 + aRow1) * kD + k0 + aC4_1);
  };

  v4f r1[2], r3[2];
  auto gloadB = [&](int k0) {
    const int r0 = k0 + bKp * 2;
    r1[0] = *(const v4f*)(B1 + (size_t)r0 * kH + n0 + bC4);
    r1[1] = *(const v4f*)(B1 + (size_t)(r0 + 1) * kH + n0 + bC4);
    r3[0] = *(const v4f*)(B3 + (size_t)r0 * kH + n0 + bC4);
    r3[1] = *(const v4f*)(B3 + (size_t)(r0 + 1) * kH + n0 + bC4);
  };
  auto lstoreB = [&]() {
#pragma unroll
    for (int j = 0; j < 4; ++j) {
      *(v2f*)&sB1[bKp][bC4 + j][0] = (v2f){r1[0][j], r1[1][j]};
      *(v2f*)&sB3[bKp][bC4 + j][0] = (v2f){r3[0][j], r3[1][j]};
    }
  };

  issueA(0, 0);
  gloadB(0);
  int buf = 0;
  for (int k0 = 0; k0 < kD; k0 += BK) {
    lstoreB();
    wait_async0();       // this wave's async A copies for `buf` are done
    __syncthreads();     // all waves' A copies + B stores visible
    if (k0 + BK < kD) {  // overlap next tile's fills with WMMA below
      issueA(buf ^ 1, k0 + BK);
      gloadB(k0 + BK);
    }

#pragma unroll
    for (int kk = 0; kk < BK; kk += 4) {
      const int kp = (kk + koff) >> 1;
      v2f a[2], b1f[4], b3f[4];
#pragma unroll
      for (int mt = 0; mt < 2; ++mt)
        a[mt] = *(const v2f*)&sA[buf][wm + mt * 16 + r16][kk + koff];
#pragma unroll
      for (int nt = 0; nt < 4; ++nt) {
        b1f[nt] = *(const v2f*)&sB1[kp][wn + nt * 16 + r16][0];
        b3f[nt] = *(const v2f*)&sB3[kp][wn + nt * 16 + r16][0];
      }
#pragma unroll
      for (int mt = 0; mt < 2; ++mt)
#pragma unroll
        for (int nt = 0; nt < 4; ++nt) {
          acc1[mt][nt] = __builtin_amdgcn_wmma_f32_16x16x4_f32(
              false, a[mt], false, b1f[nt], (short)0, acc1[mt][nt], false,
              false);
          acc3[mt][nt] = __builtin_amdgcn_wmma_f32_16x16x4_f32(
              false, a[mt], false, b3f[nt], (short)0, acc3[mt][nt], false,
              false);
        }
    }
    __syncthreads();
    buf ^= 1;
  }

  // epilogue: h = up * silu(gate)
#pragma unroll
  for (int mt = 0; mt < 2; ++mt)
#pragma unroll
    for (int nt = 0; nt < 4; ++nt)
#pragma unroll
      for (int v = 0; v < 8; ++v) {
        const int row = m0 + wm + mt * 16 + rowb + v;
        const int col = n0 + wn + nt * 16 + r16;
        const float g = acc3[mt][nt][v];
        const float silu = g / (1.0f + __expf(-g));
        Hout[(size_t)row * kH + col] = acc1[mt][nt][v] * silu;
      }
}

// ---------------------------------------------------------------------------
// Pass 2: out (+)= w_full[:,e] * (hbuf @ W2)   M=4096, N=kD, K=kH
// expert < 0  -> shared expert: weight 1, plain store (initializes out)
// ---------------------------------------------------------------------------
__global__ __launch_bounds__(256) void down_kernel(
    const float* __restrict__ A,     // [kM, kH]
    const float* __restrict__ B,     // [kH, kD]
    const float* __restrict__ Wfull, // [kM, kE]
    int expert,
    float* __restrict__ Out) {       // [kM, kD]
  __shared__ __align__(16) float sA[2][BM][BK + PAD];
  __shared__ __align__(16) float sB[BK / 2][BNP][2];

  const int m0 = blockIdx.y * BM;
  const int n0 = blockIdx.x * BN;
  const int tid = threadIdx.x;
  const int wave = tid >> 5;
  const int lane = tid & 31;
  const int wm = (wave & 3) * 32;
  const int wn = (wave >> 2) * 64;
  const int r16 = lane & 15;
  const int koff = (lane >> 4) * 2;
  const int rowb = (lane >> 4) * 8;

  const int aRow0 = tid >> 2;
  const int aC4_0 = (tid & 3) * 4;
  const int aRow1 = (tid + 256) >> 2;
  const int aC4_1 = ((tid + 256) & 3) * 4;
  const int bKp = tid >> 5;
  const int bC4 = (tid & 31) * 4;

  v8f acc[2][4];
#pragma unroll
  for (int mt = 0; mt < 2; ++mt)
#pragma unroll
    for (int nt = 0; nt < 4; ++nt) acc[mt][nt] = (v8f){0, 0, 0, 0, 0, 0, 0, 0};

  auto issueA = [&](int buf, int k0) {
    async_copy_b128(lds_addr32(&sA[buf][aRow0][aC4_0]),
                    A + (size_t)(m0 + aRow0) * kH + k0 + aC4_0);
    async_copy_b128(lds_addr32(&sA[buf][aRow1][aC4_1]),
                    A + (size_t)(m0 + aRow1) * kH + k0 + aC4_1);
  };

  v4f rbv[2];
  auto gloadB = [&](int k0) {
    const int r0 = k0 + bKp * 2;
    rbv[0] = *(const v4f*)(B + (size_t)r0 * kD + n0 + bC4);
    rbv[1] = *(const v4f*)(B + (size_t)(r0 + 1) * kD + n0 + bC4);
  };
  auto lstoreB = [&]() {
#pragma unroll
    for (int j = 0; j < 4; ++j)
      *(v2f*)&sB[bKp][bC4 + j][0] = (v2f){rbv[0][j], rbv[1][j]};
  };

  issueA(0, 0);
  gloadB(0);
  int buf = 0;
  for (int k0 = 0; k0 < kH; k0 += BK) {
    lstoreB();
    wait_async0();
    __syncthreads();
    if (k0 + BK < kH) {
      issueA(buf ^ 1, k0 + BK);
      gloadB(k0 + BK);
    }

#pragma unroll
    for (int kk = 0; kk < BK; kk += 4) {
      const int kp = (kk + koff) >> 1;
      v2f a[2], b[4];
#pragma unroll
      for (int mt = 0; mt < 2; ++mt)
        a[mt] = *(const v2f*)&sA[buf][wm + mt * 16 + r16][kk + koff];
#pragma unroll
      for (int nt = 0; nt < 4; ++nt)
        b[nt] = *(const v2f*)&sB[kp][wn + nt * 16 + r16][0];
#pragma unroll
      for (int mt = 0; mt < 2; ++mt)
#pragma unroll
        for (int nt = 0; nt < 4; ++nt)
          acc[mt][nt] = __builtin_amdgcn_wmma_f32_16x16x4_f32(
              false, a[mt], false, b[nt], (short)0, acc[mt][nt], false, false);
    }
    __syncthreads();
    buf ^= 1;
  }

#pragma unroll
  for (int mt = 0; mt < 2; ++mt)
#pragma unroll
    for (int nt = 0; nt < 4; ++nt)
#pragma unroll
      for (int v = 0; v < 8; ++v) {
        const int row = m0 + wm + mt * 16 + rowb + v;
        const int col = n0 + wn + nt * 16 + r16;
        const float val = acc[mt][nt][v];
        if (expert < 0) {
          Out[(size_t)row * kD + col] = val;  // shared expert: initialize
        } else {
          const float wtok = Wfull[(size_t)row * kE + expert];
          Out[(size_t)row * kD + col] += wtok * val;
        }
      }
}

// ---------------------------------------------------------------------------
extern "C" void kernel_launch(void* const* d_in, const int* in_sizes, int n_in,
                              void* d_out, int out_size, void* d_ws,
                              size_t ws_size, hipStream_t stream) {
  const float* x   = (const float*)d_in[0];
  const float* Wr  = (const float*)d_in[1];
  const float* rb  = (const float*)d_in[2];
  const float* W1  = (const float*)d_in[3];  // [E, D, H]
  const float* W2  = (const float*)d_in[4];  // [E, H, D]
  const float* W3  = (const float*)d_in[5];  // [E, D, H]
  const float* Ws1 = (const float*)d_in[6];  // [D, HS]
  const float* Ws2 = (const float*)d_in[7];  // [HS, D]
  const float* Ws3 = (const float*)d_in[8];  // [D, HS]
  float* out = (float*)d_out;

  // workspace layout: [w_full : kM*kE f32][hbuf : kM*kH f32]  (~33.7 MB)
  float* wfull = (float*)d_ws;
  const size_t off = ((size_t)kM * kE * sizeof(float) + 255) & ~(size_t)255;
  float* hbuf = (float*)((char*)d_ws + off);

  router_kernel<<<kM / 8, 256, 0, stream>>>(x, Wr, rb, wfull);

  dim3 blk(256, 1, 1);
  dim3 g1(kH / BN, kM / BM, 1);  // 16 x 32
  dim3 g2(kD / BN, kM / BM, 1);  // 8 x 32

  // shared expert first: initializes out
  gateup_kernel<<<g1, blk, 0, stream>>>(x, Ws1, Ws3, hbuf);
  down_kernel<<<g2, blk, 0, stream>>>(hbuf, Ws2, wfull, -1, out);

  // routed experts: accumulate with per-token router weights
  for (int e = 0; e < kE; ++e) {
    gateup_kernel<<<g1, blk, 0, stream>>>(x, W1 + (size_t)e * kD * kH,
                                          W3 + (size_t)e * kD * kH, hbuf);
    down_kernel<<<g2, blk, 0, stream>>>(hbuf, W2 + (size_t)e * kH * kD, wfull,
                                        e, out);
  }

  (void)in_sizes; (void)n_in; (void)out_size; (void)ws_size;
}